// SparseActivation_60979945669068
// MI455X (gfx1250) — compile-verified
//
#include <hip/hip_runtime.h>

#define ROW_D   2048
#define BLOCK   256
#define NWAVE   (BLOCK / 32)        // 8 waves (wave32)
#define VPT     (ROW_D / BLOCK)     // 8 floats per thread

typedef float v4f __attribute__((ext_vector_type(4)));  // native vector for NT stores

__global__ __launch_bounds__(BLOCK) void sparse_act_topk_kernel(
    const float* __restrict__ x, float* __restrict__ out,
    unsigned k, float scale)
{
    __shared__ float    s_row[ROW_D];            // 8 KB staged row
    __shared__ unsigned s_whist[NWAVE * 256];    // per-wave histograms (8 KB)
    __shared__ unsigned s_scan[256];             // reduced bin counts
    __shared__ unsigned s_prefix;
    __shared__ unsigned s_krem;

    const int t    = threadIdx.x;
    const int wid  = t >> 5;
    const int lane = t & 31;

    // ---- Stage the row HBM -> LDS with gfx1250 async-to-LDS b128 loads ----
    const unsigned long long base =
        (unsigned long long)(x + (unsigned long long)blockIdx.x * ROW_D);
    // LDS aperture: low 32 bits of a flat shared pointer are the LDS byte offset
    const unsigned ldsBase = (unsigned)(unsigned long long)(void*)s_row;
    unsigned a0 = ldsBase + (unsigned)t * 16u;
    unsigned a1 = a0 + BLOCK * 16u;
    unsigned o0 = (unsigned)t * 16u;
    unsigned o1 = o0 + BLOCK * 16u;
    asm volatile(
        "global_load_async_to_lds_b128 %0, %1, %4\n\t"
        "global_load_async_to_lds_b128 %2, %3, %4"
        :
        : "v"(a0), "v"(o0), "v"(a1), "v"(o1), "s"(base)
        : "memory");

    // zero the per-wave histograms while the async DMA is in flight
#pragma unroll
    for (int i = 0; i < NWAVE; ++i) s_whist[i * 256 + t] = 0u;

    asm volatile("s_wait_asynccnt 0" ::: "memory");
    __syncthreads();

    // pull this thread's 8 elements into registers (2x float4, bank-conflict-free)
    v4f v0 = ((const v4f*)s_row)[t];
    v4f v1 = ((const v4f*)s_row)[t + BLOCK];
    float vals[VPT] = {v0.x, v0.y, v0.z, v0.w, v1.x, v1.y, v1.z, v1.w};
    unsigned bits[VPT];
#pragma unroll
    for (int i = 0; i < VPT; ++i)
        bits[i] = __float_as_uint(vals[i]) & 0x7fffffffu;

    // ---- radix-select the k-th largest |x| bit pattern: 4 passes x 8 bits ----
    unsigned prefix = 0u, krem = k;
#pragma unroll 1
    for (int pass = 0; pass < 4; ++pass) {
        const int shift = 24 - pass * 8;

        // histogram candidates into this wave's private bins
        const unsigned himask = pass ? (0xffffffffu << (shift + 8)) : 0u;
        unsigned* myhist = &s_whist[wid * 256];
#pragma unroll
        for (int i = 0; i < VPT; ++i) {
            if ((bits[i] & himask) == (prefix & himask))
                atomicAdd(&myhist[(bits[i] >> shift) & 0xffu], 1u);
        }
        __syncthreads();

        // reduce the 8 wave-histograms (and re-zero them for the next pass:
        // same thread reads & writes column t of every histogram -> no race)
        unsigned cnt = 0u;
#pragma unroll
        for (int w = 0; w < NWAVE; ++w) {
            cnt += s_whist[w * 256 + t];
            s_whist[w * 256 + t] = 0u;
        }
        s_scan[t] = cnt;
        __syncthreads();

        // wave 0: register-resident suffix scan over 256 bins + digit select
        if (t < 32) {
            const uint4* sv = (const uint4*)s_scan;
            uint4 lo = sv[lane * 2];
            uint4 hi = sv[lane * 2 + 1];
            unsigned c[8] = {lo.x, lo.y, lo.z, lo.w, hi.x, hi.y, hi.z, hi.w};
            unsigned mySum = 0u;
#pragma unroll
            for (int j = 0; j < 8; ++j) mySum += c[j];
            // inclusive suffix sum across lanes (5 shfl steps, no barriers)
            unsigned acc = mySum;
#pragma unroll
            for (int o = 1; o < 32; o <<= 1) {
                unsigned v = __shfl_down(acc, o, 32);
                if (lane + o < 32) acc += v;
            }
            const unsigned above = acc - mySum;  // suffix of bin (lane*8 + 8)
            unsigned sufnext = above;
#pragma unroll
            for (int j = 7; j >= 0; --j) {
                unsigned suf = sufnext + c[j];    // suffix of bin lane*8+j
                if (suf >= krem && sufnext < krem) {   // unique crossing
                    s_prefix = prefix | ((unsigned)(lane * 8 + j) << shift);
                    s_krem   = krem - sufnext;
                }
                sufnext = suf;
            }
        }
        __syncthreads();
        prefix = s_prefix;
        krem   = s_krem;
    }

    // ---- apply mask + scale, coalesced non-temporal b128 stores ----
    const unsigned thr = prefix;   // bit pattern of the k-th largest |x|
    v4f r0, r1;
    r0.x = (bits[0] >= thr) ? vals[0] * scale : 0.0f;
    r0.y = (bits[1] >= thr) ? vals[1] * scale : 0.0f;
    r0.z = (bits[2] >= thr) ? vals[2] * scale : 0.0f;
    r0.w = (bits[3] >= thr) ? vals[3] * scale : 0.0f;
    r1.x = (bits[4] >= thr) ? vals[4] * scale : 0.0f;
    r1.y = (bits[5] >= thr) ? vals[5] * scale : 0.0f;
    r1.z = (bits[6] >= thr) ? vals[6] * scale : 0.0f;
    r1.w = (bits[7] >= thr) ? vals[7] * scale : 0.0f;

    v4f* orow = (v4f*)(out + (unsigned long long)blockIdx.x * ROW_D);
    __builtin_nontemporal_store(r0, orow + t);
    __builtin_nontemporal_store(r1, orow + t + BLOCK);
}

extern "C" void kernel_launch(void* const* d_in, const int* in_sizes, int n_in,
                              void* d_out, int out_size, void* d_ws, size_t ws_size,
                              hipStream_t stream) {
    const float* x   = (const float*)d_in[0];
    float*       out = (float*)d_out;

    const long long total = (long long)in_sizes[0];       // B*S*D
    const int rows = (int)(total / ROW_D);                // 16384 rows
    unsigned k = ROW_D / 10; if (k < 1u) k = 1u;          // == int(D*0.1) = 204
    const float scale = (float)ROW_D / (float)k;

    hipLaunchKernelGGL(sparse_act_topk_kernel,
                       dim3(rows), dim3(BLOCK), 0, stream,
                       x, out, k, scale);
}